// ReEig_50577534878058
// MI455X (gfx1250) — compile-verified
//
#include <hip/hip_runtime.h>
#include <hip/hip_bf16.h>

// ReEig for 8192 symmetric 64x64 f32 matrices on gfx1250 (MI455X).
// Symmetric SVD == eigendecomposition: Out = Q * diag(sign(l)*max(|l|,eps)) * Q^T.
// Phase 1: batched parallel-order cyclic Jacobi eigensolver in LDS (f32 VALU).
// Phase 2: reconstruction GEMM via V_WMMA_F32_16X16X4_F32 (wave32 WMMA).

#define MATN     64
#define LDSP     65      // LDS row stride in floats: 65 mod 64 == 1 -> conflict-free column walks
#define NTHREADS 256     // 8 wave32 per block
#define SWEEPS   8
#define EPSILON  1e-4f

typedef __attribute__((ext_vector_type(2))) float v2f;
typedef __attribute__((ext_vector_type(8))) float v8f;

__global__ __launch_bounds__(NTHREADS)
void reeig_jacobi_wmma_kernel(const float* __restrict__ in,
                              float* __restrict__ out)
{
    __shared__ float A[MATN * LDSP];   // working symmetric matrix
    __shared__ float Q[MATN * LDSP];   // accumulated eigenvectors
    __shared__ float rot_c[32];
    __shared__ float rot_s[32];
    __shared__ int   rot_p[32];
    __shared__ int   rot_q[32];
    __shared__ float fval[MATN];       // clamped eigenvalues

    const int tid = threadIdx.x;
    const int b   = blockIdx.x;
    const float* __restrict__ src = in  + (size_t)b * (MATN * MATN);
    float*       __restrict__ dst = out + (size_t)b * (MATN * MATN);

    // ---- Stage input into LDS, init Q = I ----------------------------------
    for (int i = tid; i < MATN * MATN; i += NTHREADS) {
        const int r = i >> 6;
        const int c = i & 63;
        A[r * LDSP + c] = src[i];
        Q[r * LDSP + c] = (r == c) ? 1.0f : 0.0f;
    }
    __syncthreads();

    // ---- Parallel-order cyclic Jacobi --------------------------------------
    // Round-robin tournament: 63 rounds/sweep, 32 disjoint pairs/round covers
    // every (p,q) pair exactly once per sweep. Disjoint rotations commute, so
    // compute all angles from a snapshot, then apply all column updates, then
    // all row updates.
    for (int sweep = 0; sweep < SWEEPS; ++sweep) {
        for (int round = 0; round < 63; ++round) {
            // Phase 1: angles for the 32 disjoint pairs (wave 0).
            if (tid < 32) {
                int p, q;
                if (tid == 0) { p = 63; q = round; }
                else {
                    p = (round + tid) % 63;
                    q = (round + 63 - tid) % 63;
                }
                if (p > q) { const int t = p; p = q; q = t; }
                const float apq = A[p * LDSP + q];
                float c = 1.0f, s = 0.0f;
                if (__builtin_fabsf(apq) > 1e-12f) {
                    const float app = A[p * LDSP + p];
                    const float aqq = A[q * LDSP + q];
                    const float tau = (aqq - app) / (2.0f * apq);
                    const float t   = __builtin_copysignf(1.0f, tau) /
                                      (__builtin_fabsf(tau) + sqrtf(1.0f + tau * tau));
                    c = rsqrtf(1.0f + t * t);
                    s = t * c;
                }
                rot_p[tid] = p; rot_q[tid] = q;
                rot_c[tid] = c; rot_s[tid] = s;
            }
            __syncthreads();

            // Phase 2: column updates  A <- A*J,  Q <- Q*J   (32 pairs x 64 rows)
            for (int idx = tid; idx < 32 * MATN; idx += NTHREADS) {
                const int j = idx >> 6;
                const int k = idx & 63;
                const int p = rot_p[j], q = rot_q[j];
                const float c = rot_c[j], s = rot_s[j];
                const float akp = A[k * LDSP + p];
                const float akq = A[k * LDSP + q];
                A[k * LDSP + p] = fmaf(c, akp, -s * akq);
                A[k * LDSP + q] = fmaf(s, akp,  c * akq);
                const float qkp = Q[k * LDSP + p];
                const float qkq = Q[k * LDSP + q];
                Q[k * LDSP + p] = fmaf(c, qkp, -s * qkq);
                Q[k * LDSP + q] = fmaf(s, qkp,  c * qkq);
            }
            __syncthreads();

            // Phase 3: row updates  A <- J^T * A
            for (int idx = tid; idx < 32 * MATN; idx += NTHREADS) {
                const int j = idx >> 6;
                const int k = idx & 63;
                const int p = rot_p[j], q = rot_q[j];
                const float c = rot_c[j], s = rot_s[j];
                const float apk = A[p * LDSP + k];
                const float aqk = A[q * LDSP + k];
                A[p * LDSP + k] = fmaf(c, apk, -s * aqk);
                A[q * LDSP + k] = fmaf(s, apk,  c * aqk);
            }
            __syncthreads();
        }
    }

    // ---- ReEig spectral clamp: f(l) = sign(l) * max(|l|, eps) --------------
    if (tid < MATN) {
        const float lam = A[tid * LDSP + tid];
        fval[tid] = __builtin_copysignf(fmaxf(__builtin_fabsf(lam), EPSILON), lam);
    }
    __syncthreads();

    // ---- Reconstruction: Out = (Q * diag(f)) * Q^T via FP32 WMMA -----------
    // 16 output tiles of 16x16; 8 waves -> 2 tiles per wave. Each tile
    // accumulates 16 K=4 steps with V_WMMA_F32_16X16X4_F32.
    //
    // Layouts (ISA 7.12.2, wave32):
    //   A 16x4  : lane L holds row m=(L&15); VGPR0 = K{0 | 2}, VGPR1 = K{1 | 3}
    //             (K offset 2*(L>>4)).
    //   B 4x16  : lane L holds col n=(L&15); VGPR0 = K{0 | 2}, VGPR1 = K{1 | 3}.
    //   C/D     : VGPR v holds rows M = v + 8*(L>>4), N = (L&15).
    const int lane  = tid & 31;
    const int wave  = tid >> 5;
    const int mlane = lane & 15;
    const int khalf = lane >> 4;

    for (int t = wave; t < 16; t += 8) {            // uniform across all waves
        const int tm = (t >> 2) * 16;               // output row tile base
        const int tn = (t & 3) * 16;                // output col tile base
        v8f acc = {};
#pragma unroll
        for (int kk = 0; kk < 16; ++kk) {
            const int k0 = kk * 4 + khalf * 2;      // this lane's K offset
            v2f afrag, bfrag;
            // A = Q scaled by f along columns:  A[m][k] = Q[m][k] * f[k]
            afrag.x = Q[(tm + mlane) * LDSP + k0    ] * fval[k0    ];
            afrag.y = Q[(tm + mlane) * LDSP + k0 + 1] * fval[k0 + 1];
            // B = Q^T:  B[k][n] = Q[n][k]
            bfrag.x = Q[(tn + mlane) * LDSP + k0    ];
            bfrag.y = Q[(tn + mlane) * LDSP + k0 + 1];
            acc = __builtin_amdgcn_wmma_f32_16x16x4_f32(
                      /*neg_a=*/false, afrag,
                      /*neg_b=*/false, bfrag,
                      /*c_mod=*/(short)0, acc,
                      /*reuse_a=*/false, /*reuse_b=*/false);
        }
#pragma unroll
        for (int v = 0; v < 8; ++v) {
            dst[(tm + v + 8 * khalf) * MATN + tn + mlane] = acc[v];
        }
    }
}

extern "C" void kernel_launch(void* const* d_in, const int* in_sizes, int n_in,
                              void* d_out, int out_size, void* d_ws, size_t ws_size,
                              hipStream_t stream) {
    (void)n_in; (void)out_size; (void)d_ws; (void)ws_size;
    const float* in  = (const float*)d_in[0];
    float*       out = (float*)d_out;
    const int nmat = in_sizes[0] / (MATN * MATN);   // 8192 for the reference shapes
    reeig_jacobi_wmma_kernel<<<nmat, NTHREADS, 0, stream>>>(in, out);
}